// StateTrajectoryNetwork_29326036697653
// MI455X (gfx1250) — compile-verified
//
#include <hip/hip_runtime.h>
#include <hip/hip_bf16.h>

typedef float v2f __attribute__((ext_vector_type(2)));
typedef float v8f __attribute__((ext_vector_type(8)));

#if __has_builtin(__builtin_amdgcn_tanhf)
#define TANHF(v) __builtin_amdgcn_tanhf(v)
#else
#define TANHF(v) tanhf(v)
#endif

#define S_STEPS 4096
#define BATCH   16384
#define PF_DIST 16

// Cross half-wave swap (lane <-> lane^16) as pure VALU: v_permlanex16_b32
// with identity lane selects. Avoids the ds_bpermute + s_wait_dscnt round
// trip that sits on the serial recurrence chain.
static __device__ __forceinline__ float swap16(float v) {
#if __has_builtin(__builtin_amdgcn_permlanex16)
    int i = __float_as_int(v);
    i = __builtin_amdgcn_permlanex16(i, i, 0x76543210u, 0xfedcba98u, false, false);
    return __int_as_float(i);
#else
    return __shfl_xor(v, 16, 32);
#endif
}

// One wave (32 lanes) owns 16 batch columns. Per step, 4 chained
// v_wmma_f32_16x16x4_f32 compute the whole MLP for those 16 columns,
// with the recurrent state living in lanes 0-15.
__global__ __launch_bounds__(32)
void stn_rnn_wmma_kernel(const float* __restrict__ x,
                         const float* __restrict__ W1,
                         const float* __restrict__ W2,
                         const float* __restrict__ b2,
                         const float* __restrict__ W3,
                         const float* __restrict__ W4,
                         float* __restrict__ out)
{
    const int  lane = threadIdx.x;        // 0..31 (wave32)
    const int  col  = lane & 15;          // batch column within tile
    const bool lo   = lane < 16;
    const size_t base = (size_t)blockIdx.x * 16;   // 1024 tiles of 16 columns

    // ---- A-matrix fragments (16x4 f32, ISA layout):
    // lanes 0-15 : row M=lane, {K0,K1}; lanes 16-31 : row M=lane-16, {K2,K3}.
    // Weight matrices occupy rows 0-3; all other rows are zero.
    const int  r      = col;
    const bool row_ok = (r < 4);

    v2f a1 = {0.f, 0.f}, a2 = {0.f, 0.f}, a3 = {0.f, 0.f}, a4 = {0.f, 0.f};
    if (row_ok) {
        if (lo) { a1.x = W1[r*2 + 0]; a1.y = W1[r*2 + 1]; }   // W1 is 4x2; K2/K3 = 0
        a2.x = W2[r*4 + (lo ? 0 : 2)]; a2.y = W2[r*4 + (lo ? 1 : 3)];
        a3.x = W3[r*4 + (lo ? 0 : 2)]; a3.y = W3[r*4 + (lo ? 1 : 3)];
    }
    if (r == 0) {                                             // A4 = 2*W4 in row 0
        a4.x = 2.0f * W4[lo ? 0 : 2];
        a4.y = 2.0f * W4[lo ? 1 : 3];
    }

    // C fragment: bias for layer 2 (rows 0-3, lanes 0-15 hold M=0..7).
    v8f czero = {};
    v8f cbias = {};
    cbias[0] = lo ? b2[0] : 0.f;
    cbias[1] = lo ? b2[1] : 0.f;
    cbias[2] = lo ? b2[2] : 0.f;
    cbias[3] = lo ? b2[3] : 0.f;

    float s = 0.f;                         // recurrent state (valid in lanes 0-15)
    const float* xp = x   + base;
    float*       op = out + base;

    // Software pipeline: keep next timestep's x in flight while the serial
    // WMMA chain of the current step executes.
    float xcur = __builtin_nontemporal_load(xp + col);

    for (int t = 0; t < S_STEPS; ++t) {
        const size_t idx = (size_t)t * BATCH + col;

        if (t + PF_DIST < S_STEPS)
            __builtin_prefetch(xp + idx + (size_t)PF_DIST * BATCH, 0, 0);

        float xnext = 0.f;
        if (t + 1 < S_STEPS)
            xnext = __builtin_nontemporal_load(xp + idx + BATCH);

        // ---- layer 1: D1 = W1 @ [x; s]
        v2f bm;
        bm.x = lo ? xcur : 0.f;            // K0 = x_t
        bm.y = lo ? s    : 0.f;            // K1 = s
        v8f d = __builtin_amdgcn_wmma_f32_16x16x4_f32(
                    false, a1, false, bm, (short)0, czero, false, false);

        float t0 = TANHF(d[0]), t1 = TANHF(d[1]), t2 = TANHF(d[2]), t3 = TANHF(d[3]);
        float t2s = swap16(t2), t3s = swap16(t3);
        bm.x = lo ? t0 : t2s;              // rows 2,3 become K2,K3 in hi lanes
        bm.y = lo ? t1 : t3s;

        // ---- layer 2: D2 = W2 @ h1 + b2
        d = __builtin_amdgcn_wmma_f32_16x16x4_f32(
                false, a2, false, bm, (short)0, cbias, false, false);
        t0 = TANHF(d[0]); t1 = TANHF(d[1]); t2 = TANHF(d[2]); t3 = TANHF(d[3]);
        t2s = swap16(t2); t3s = swap16(t3);
        bm.x = lo ? t0 : t2s;
        bm.y = lo ? t1 : t3s;

        // ---- layer 3: D3 = W3 @ h2
        d = __builtin_amdgcn_wmma_f32_16x16x4_f32(
                false, a3, false, bm, (short)0, czero, false, false);
        t0 = TANHF(d[0]); t1 = TANHF(d[1]); t2 = TANHF(d[2]); t3 = TANHF(d[3]);
        t2s = swap16(t2); t3s = swap16(t3);
        bm.x = lo ? t0 : t2s;
        bm.y = lo ? t1 : t3s;

        // ---- layer 4 (residual): s' = s + (2*W4) @ h3   (C = inline 0)
        d = __builtin_amdgcn_wmma_f32_16x16x4_f32(
                false, a4, false, bm, (short)0, czero, false, false);
        s = s + d[0];                      // row 0, lanes 0-15 = s_{t+1} = y_t

        if (lo)
            __builtin_nontemporal_store(s, op + idx);

        xcur = xnext;
    }
}

extern "C" void kernel_launch(void* const* d_in, const int* in_sizes, int n_in,
                              void* d_out, int out_size, void* d_ws, size_t ws_size,
                              hipStream_t stream) {
    const float* x  = (const float*)d_in[0];
    const float* W1 = (const float*)d_in[1];
    const float* W2 = (const float*)d_in[2];
    const float* b2 = (const float*)d_in[3];
    const float* W3 = (const float*)d_in[4];
    const float* W4 = (const float*)d_in[5];

    dim3 grid(BATCH / 16);   // 1024 single-wave workgroups -> fills every SIMD
    dim3 block(32);
    stn_rnn_wmma_kernel<<<grid, block, 0, stream>>>(
        x, W1, W2, b2, W3, W4, (float*)d_out);
}